// GMM_64003602645506
// MI455X (gfx1250) — compile-verified
//
#include <hip/hip_runtime.h>
#include <hip/hip_bf16.h>
#include <stdint.h>

// ---------------------------------------------------------------------------
// GMM E-step negative log-likelihood on gfx1250 (MI455X).
//
// maha via z = [Linv | -Linv*mu] @ [x;1]:  batched GEMM M=2048, K=36, N=100000
// executed with V_WMMA_F32_16X16X4_F32 (full fp32 on the matrix pipe).
// Aaug (64 comps x 32 rows x 36 cols, f32) staged into LDS in two 144KB
// chunks using GLOBAL_LOAD_ASYNC_TO_LDS_B128 (ASYNCcnt-tracked direct-to-LDS
// copy, no VGPR round trip). Y never materialized: squared-sums and
// exp-weighted accumulators fused in registers.
// ---------------------------------------------------------------------------

typedef __attribute__((ext_vector_type(2))) float v2f;
typedef __attribute__((ext_vector_type(8))) float v8f;

#define NS 100000          // N_SAMPLES
#define NC 64              // N_COMPONENTS
#define ND 32              // N_FEATURES
#define ROWSTRIDE 36       // 32 cols + [-c | 0 0 0] augmentation
#define CHUNK_COMPS 32
#define CHUNK_FLOATS (CHUNK_COMPS * 32 * ROWSTRIDE)   // 36864 floats = 144 KB
#define LDS_FLOATS (CHUNK_FLOATS + 128)               // + hld[64] + w[64]
#define N_TILES ((NS + 15) / 16)                      // 6250 (exact)
#define WAVES_PER_WG 8
#define N_WG ((N_TILES + WAVES_PER_WG - 1) / WAVES_PER_WG)  // 782

#define AAUG_FLOATS (NC * 32 * ROWSTRIDE)             // 73728
#define LOG2PI_X32 58.81206612509737f                 // 32 * log(2*pi)

// ---------------------------------------------------------------------------
// Phase 0: per component k (one wave each): A = inv(L) by forward substitution
// (thread j solves column j, fully unrolled -> registers), c = A*mu,
// write Aaug row-major [k*32+i][0..35] = [A_row_i | -c_i | 0 0 0],
// half-logdet and softmax weights.
// ---------------------------------------------------------------------------
__global__ __launch_bounds__(32) void gmm_precompute(
    const float* __restrict__ means,
    const float* __restrict__ scale_tril,
    const float* __restrict__ log_weights,
    float* __restrict__ aaug,
    float* __restrict__ hldw)   // hldw[0..63]=half_logdet, [64..127]=weights
{
  const int k = blockIdx.x;
  const int j = threadIdx.x;                 // column owned by this thread
  const float* L = scale_tril + (size_t)k * ND * ND;

  float a[ND];
  #pragma unroll
  for (int i = 0; i < ND; ++i) {
    float s = (i == j) ? 1.0f : 0.0f;
    #pragma unroll
    for (int t = 0; t < ND; ++t) {
      if (t < i) s -= L[i * ND + t] * a[t];  // a[t]==0 for t<j
    }
    a[i] = s / L[i * ND + i];
  }

  __shared__ float As[ND * 33];              // 33 stride: bank-conflict-free
  #pragma unroll
  for (int i = 0; i < ND; ++i) As[i * 33 + j] = a[i];
  __syncthreads();

  // thread r owns output row r
  const int r = j;
  float c = 0.0f;
  #pragma unroll
  for (int t = 0; t < ND; ++t) c += As[r * 33 + t] * means[k * ND + t];

  float* dst = aaug + (size_t)(k * 32 + r) * ROWSTRIDE;
  #pragma unroll
  for (int t = 0; t < ND; ++t) dst[t] = As[r * 33 + t];
  dst[32] = -c;
  dst[33] = 0.0f; dst[34] = 0.0f; dst[35] = 0.0f;

  // half log-determinant (wave32 reduction)
  float hl = __logf(L[r * ND + r]);
  #pragma unroll
  for (int off = 16; off > 0; off >>= 1) hl += __shfl_down(hl, off, 32);
  if (r == 0) hldw[k] = hl;

  // softmax weights (redundant per block, thread 0 writes)
  if (r == 0) {
    float mx = -1e30f;
    for (int t = 0; t < NC; ++t) mx = fmaxf(mx, log_weights[t]);
    float den = 0.0f;
    for (int t = 0; t < NC; ++t) den += __expf(log_weights[t] - mx);
    hldw[NC + k] = __expf(log_weights[k] - mx) / den;
  }
}

// ---------------------------------------------------------------------------
// Main kernel: 8 waves/WG, each wave owns a 16-sample strip.
// Per component: 2 M-tiles x 9 K-steps of V_WMMA_F32_16X16X4_F32,
// fused maha -> logp -> running (S1, S2) so loglik = S2/S1.
// ---------------------------------------------------------------------------
__global__ __launch_bounds__(256) void gmm_estep(
    const float* __restrict__ X,
    const float* __restrict__ aaug,
    const float* __restrict__ hldw,
    float* __restrict__ partials)
{
  extern __shared__ float smem[];            // [CHUNK_FLOATS] Aaug | [128] hld,w
  const int tid  = threadIdx.x;
  const int lane = tid & 31;
  const int wave = tid >> 5;
  const int h    = lane >> 4;                // lane half (K split)
  const int r    = lane & 15;                // M / N index within tile

  if (tid < 128) smem[CHUNK_FLOATS + tid] = hldw[tid];   // broadcast tables

  // warm L2 with the second Aaug chunk while chunk 0 streams in
  __builtin_prefetch(aaug + CHUNK_FLOATS + (size_t)tid * 144, 0, 1);

  const int  tile = blockIdx.x * WAVES_PER_WG + wave;
  const long base = (long)tile * 16;
  long row = base + r; if (row > NS - 1) row = NS - 1;   // clamp (EXEC stays full)

  // B operand: lane (h, n) holds X[n][4j+2h, 4j+2h+1] per K-window j
  const float2* xr = (const float2*)(X + row * ND);
  float xb[18];
  #pragma unroll
  for (int jj = 0; jj < 8; ++jj) {
    float2 p = xr[jj * 2 + h];
    xb[2 * jj]     = p.x;
    xb[2 * jj + 1] = p.y;
  }
  xb[16] = (h == 0) ? 1.0f : 0.0f;           // augmented column: x_aug = 1
  xb[17] = 0.0f;

  // LDS byte address of smem: low 32 bits of the flat shared pointer
  const unsigned ldsbase = (unsigned)(uintptr_t)(void*)smem;

  float S1 = 0.0f, S2 = 0.0f;                // sum w*e^lp, sum w*e^lp*lp

  for (int chunk = 0; chunk < 2; ++chunk) {
    __syncthreads();
    {   // 144KB chunk: 36 x b128 per thread, direct global->LDS async copy
      const char* src = (const char*)(aaug + chunk * CHUNK_FLOATS);
      #pragma unroll
      for (int t = 0; t < 36; ++t) {
        unsigned boff = (unsigned)((t * 256 + tid) * 16);  // byte offset
        unsigned ldsa = ldsbase + boff;
        asm volatile("global_load_async_to_lds_b128 %0, %1, %2"
                     :: "v"(ldsa), "v"(boff), "s"(src)
                     : "memory");
      }
      asm volatile("s_wait_asynccnt 0" ::: "memory");
    }
    __syncthreads();

    for (int kc = 0; kc < CHUNK_COMPS; ++kc) {
      const int k = chunk * CHUNK_COMPS + kc;
      const float* A0 = &smem[(kc * 32 + r) * ROWSTRIDE + 2 * h];       // rows 0..15
      const float* A1 = A0 + 16 * ROWSTRIDE;                            // rows 16..31

      v8f acc0 = {0.f,0.f,0.f,0.f,0.f,0.f,0.f,0.f};
      v8f acc1 = {0.f,0.f,0.f,0.f,0.f,0.f,0.f,0.f};
      #pragma unroll
      for (int jj = 0; jj < 9; ++jj) {       // K = 36 in 9 steps of 4
        v2f a0 = *(const v2f*)(A0 + 4 * jj);
        v2f a1 = *(const v2f*)(A1 + 4 * jj);
        v2f b;  b.x = xb[2 * jj];  b.y = xb[2 * jj + 1];
        acc0 = __builtin_amdgcn_wmma_f32_16x16x4_f32(
                   false, a0, false, b, (short)0, acc0, false, false);
        acc1 = __builtin_amdgcn_wmma_f32_16x16x4_f32(
                   false, a1, false, b, (short)0, acc1, false, false);
      }

      // maha: this lane covers rows {8h..8h+7} U {16+8h..16+8h+7}; partner
      // lane (lane^16) covers the complement -> one shuffle completes it.
      float part = 0.0f;
      #pragma unroll
      for (int v = 0; v < 8; ++v) part += acc0[v] * acc0[v] + acc1[v] * acc1[v];
      part += __shfl_xor(part, 16, 32);

      const float lp = -0.5f * (LOG2PI_X32 + part) - smem[CHUNK_FLOATS + k];
      const float e  = smem[CHUNK_FLOATS + NC + k] * __expf(lp);
      S1 += e;
      S2 += e * lp;
    }
  }

  // per-sample loglik = S2/S1 (valid in lanes 0..15; lanes 16..31 duplicate)
  float contrib = (h == 0 && (base + r) < NS) ? (S2 / S1) : 0.0f;
  #pragma unroll
  for (int off = 16; off > 0; off >>= 1) contrib += __shfl_down(contrib, off, 32);

  __syncthreads();
  if (lane == 0) smem[wave] = contrib;
  __syncthreads();
  if (tid == 0) {
    float s = 0.0f;
    #pragma unroll
    for (int w = 0; w < WAVES_PER_WG; ++w) s += smem[w];
    partials[blockIdx.x] = s;
  }
}

// ---------------------------------------------------------------------------
// Deterministic finalize: serial fixed-order sum of 782 partials (no float
// atomics -> bitwise-stable across graph replays), out = -sum/N.
// ---------------------------------------------------------------------------
__global__ void gmm_finalize(const float* __restrict__ partials, int n,
                             float* __restrict__ out) {
  if (threadIdx.x == 0 && blockIdx.x == 0) {
    float s = 0.0f;
    for (int i = 0; i < n; ++i) s += partials[i];
    out[0] = -s / (float)NS;
  }
}

extern "C" void kernel_launch(void* const* d_in, const int* in_sizes, int n_in,
                              void* d_out, int out_size, void* d_ws, size_t ws_size,
                              hipStream_t stream) {
  const float* X     = (const float*)d_in[0];   // [100000, 32]
  const float* means = (const float*)d_in[1];   // [64, 32]
  const float* Ltri  = (const float*)d_in[2];   // [64, 32, 32]
  const float* lw    = (const float*)d_in[3];   // [64]

  float* ws       = (float*)d_ws;
  float* aaug     = ws;                          // 73728 floats (288 KB)
  float* hldw     = ws + AAUG_FLOATS;            // 128 floats
  float* partials = ws + AAUG_FLOATS + 128;      // N_WG floats

  // allow >64KB dynamic LDS (host-side attribute; idempotent, capture-safe)
  (void)hipFuncSetAttribute((const void*)gmm_estep,
                            hipFuncAttributeMaxDynamicSharedMemorySize,
                            (int)(LDS_FLOATS * sizeof(float)));

  gmm_precompute<<<NC, 32, 0, stream>>>(means, Ltri, lw, aaug, hldw);
  gmm_estep<<<N_WG, 256, LDS_FLOATS * sizeof(float), stream>>>(X, aaug, hldw,
                                                               partials);
  gmm_finalize<<<1, 32, 0, stream>>>(partials, N_WG, (float*)d_out);
}